// TimeGAT_12575664243037
// MI455X (gfx1250) — compile-verified
//
#include <hip/hip_runtime.h>

typedef __attribute__((ext_vector_type(2))) float v2f;
typedef __attribute__((ext_vector_type(8))) float v8f;

#define GAT_B 16
#define GAT_K 128
#define GAT_F 128
#define GAT_H 256
#define GAT_ALPHA 0.2f

// ---------------------------------------------------------------------------
// Kernel 1: s1 = v @ W1^T, s2 = v @ W2^T   (v[b] = x[b]^T, (K,F))
// One wave per 16x16 output tile; fp32 WMMA, K-step 4, 32 steps over F=128.
// A 16x4 layout (ISA 7.12.2): lanes 0-15 rows M, VGPR0 = K{0|2}, VGPR1 = K{1|3}.
// B 4x16 dual layout: VGPR0 = row K{0|2} striped over lanes, VGPR1 = K{1|3}.
// C: VGPR r -> M = r (lanes 0-15) / r+8 (lanes 16-31), N = lane&15.
// ---------------------------------------------------------------------------
__global__ void __launch_bounds__(32)
gat_qk_gemm(const float* __restrict__ x,      // (B,F,K)
            const float* __restrict__ w,      // (H, 2F)
            float* __restrict__ s1,           // (B,K,H)
            float* __restrict__ s2) {         // (B,K,H)
  int tile = blockIdx.x;
  const int ht = tile % (GAT_H / 16); tile /= (GAT_H / 16);
  const int it = tile % (GAT_K / 16); tile /= (GAT_K / 16);
  const int b  = tile;
  const int i0 = it * 16, h0 = ht * 16;
  const int lane = threadIdx.x & 31;
  const int hf   = lane >> 4;     // 0: K pair {0,1}, 1: K pair {2,3}
  const int l    = lane & 15;

  const float* xb = x + b * GAT_F * GAT_K;
  const float* wr = w + (h0 + l) * (2 * GAT_F);

  v8f c1 = {};
  v8f c2 = {};
  for (int f0 = 0; f0 < GAT_F; f0 += 4) {
    v2f av, b1, b2;
    // A[m][k] = v[i0+m][f0+k] = x[b][f0+k][i0+m]  (coalesced across lanes)
    av.x = xb[(f0 + 2 * hf + 0) * GAT_K + i0 + l];
    av.y = xb[(f0 + 2 * hf + 1) * GAT_K + i0 + l];
    // B[k][n] = W1[h0+n][f0+k]
    b1.x = wr[f0 + 2 * hf + 0];
    b1.y = wr[f0 + 2 * hf + 1];
    b2.x = wr[GAT_F + f0 + 2 * hf + 0];
    b2.y = wr[GAT_F + f0 + 2 * hf + 1];
    c1 = __builtin_amdgcn_wmma_f32_16x16x4_f32(false, av, false, b1, (short)0,
                                               c1, false, false);
    c2 = __builtin_amdgcn_wmma_f32_16x16x4_f32(false, av, false, b2, (short)0,
                                               c2, false, false);
  }
#pragma unroll
  for (int r = 0; r < 8; ++r) {
    const int row = i0 + r + 8 * hf;
    const int col = h0 + l;
    s1[(b * GAT_K + row) * GAT_H + col] = c1[r];
    s2[(b * GAT_K + row) * GAT_H + col] = c2[r];
  }
}

// ---------------------------------------------------------------------------
// Kernel 2: e[b,i,j] = sum_h lrelu(s1[b,i,h] + s2[b,j,h] + lb[h]) * a[h]
//                     + bias[i,j]; attn = softmax_j(e).
// Block = (b, 8 rows), 256 threads (8 waves, 1 row each). s2 staged in LDS in
// two 64-row / 64 KB chunks. Lanes own h (h = lane + 32r), butterfly-reduce
// per j; softmax held in registers (j = lane + 32t).
// ---------------------------------------------------------------------------
__global__ void __launch_bounds__(256)
gat_attn(const float* __restrict__ s1,
         const float* __restrict__ s2,
         const float* __restrict__ lb,     // (H)
         const float* __restrict__ av,     // (H)
         const float* __restrict__ bias,   // (K,K)
         float* __restrict__ attn) {       // (B,K,K)
  __shared__ float s2lds[64 * GAT_H];      // 64 KB

  const int blk  = blockIdx.x;
  const int ic   = blk % (GAT_K / 8);
  const int b    = blk / (GAT_K / 8);
  const int wave = threadIdx.x >> 5;
  const int lane = threadIdx.x & 31;
  const int i    = ic * 8 + wave;

  // s1 row with linear_b folded in; attention vector a. h = lane + 32r.
  float r1[8], ar[8];
#pragma unroll
  for (int r = 0; r < 8; ++r) {
    const int h = lane + 32 * r;
    r1[r] = s1[(b * GAT_K + i) * GAT_H + h] + lb[h];
    ar[r] = av[h];
  }

  float e0 = 0.f, e1 = 0.f, e2 = 0.f, e3 = 0.f;   // e[j = lane + 32t]

#pragma unroll
  for (int c = 0; c < 2; ++c) {
    __syncthreads();
    for (int idx = threadIdx.x; idx < 64 * GAT_H; idx += 256) {
      const int jj = idx >> 8;           // 0..63
      const int h  = idx & (GAT_H - 1);  // 0..255
      s2lds[idx] = s2[(b * GAT_K + c * 64 + jj) * GAT_H + h];
    }
    __syncthreads();

#pragma unroll
    for (int t2 = 0; t2 < 2; ++t2) {
      for (int jl = 0; jl < 32; ++jl) {
        const int jj = t2 * 32 + jl;
        float acc = 0.f;
#pragma unroll
        for (int r = 0; r < 8; ++r) {
          float t = r1[r] + s2lds[jj * GAT_H + lane + 32 * r];
          t = fmaxf(t, GAT_ALPHA * t);   // leaky_relu, alpha > 0
          acc = fmaf(t, ar[r], acc);
        }
        acc += __shfl_xor(acc, 1);
        acc += __shfl_xor(acc, 2);
        acc += __shfl_xor(acc, 4);
        acc += __shfl_xor(acc, 8);
        acc += __shfl_xor(acc, 16);
        const int j = c * 64 + jj;
        acc += bias[i * GAT_K + j];
        if (jl == lane) {
          if (c * 2 + t2 == 0)      e0 = acc;
          else if (c * 2 + t2 == 1) e1 = acc;
          else if (c * 2 + t2 == 2) e2 = acc;
          else                      e3 = acc;
        }
      }
    }
  }

  // softmax over the 128 j values distributed as e0..e3 across 32 lanes
  float m = fmaxf(fmaxf(e0, e1), fmaxf(e2, e3));
  m = fmaxf(m, __shfl_xor(m, 1));
  m = fmaxf(m, __shfl_xor(m, 2));
  m = fmaxf(m, __shfl_xor(m, 4));
  m = fmaxf(m, __shfl_xor(m, 8));
  m = fmaxf(m, __shfl_xor(m, 16));
  e0 = __expf(e0 - m);
  e1 = __expf(e1 - m);
  e2 = __expf(e2 - m);
  e3 = __expf(e3 - m);
  float s = e0 + e1 + e2 + e3;
  s += __shfl_xor(s, 1);
  s += __shfl_xor(s, 2);
  s += __shfl_xor(s, 4);
  s += __shfl_xor(s, 8);
  s += __shfl_xor(s, 16);
  const float inv = 1.0f / s;
  float* arow = attn + (b * GAT_K + i) * GAT_K;
  arow[lane +  0] = e0 * inv;
  arow[lane + 32] = e1 * inv;
  arow[lane + 64] = e2 * inv;
  arow[lane + 96] = e3 * inv;
}

// ---------------------------------------------------------------------------
// Kernel 3: out = sigmoid(attn[b] @ v[b]); fp32 WMMA, fused sigmoid.
// B[k][n] = v[b][j0+k][f0+n] = x[b][f0+n][j0+k].
// ---------------------------------------------------------------------------
__global__ void __launch_bounds__(32)
gat_out(const float* __restrict__ attn,    // (B,K,K)
        const float* __restrict__ x,       // (B,F,K)
        float* __restrict__ out) {         // (B,K,F)
  int tile = blockIdx.x;
  const int ft = tile % (GAT_F / 16); tile /= (GAT_F / 16);
  const int it = tile % (GAT_K / 16); tile /= (GAT_K / 16);
  const int b  = tile;
  const int i0 = it * 16, f0 = ft * 16;
  const int lane = threadIdx.x & 31;
  const int hf   = lane >> 4;
  const int l    = lane & 15;

  const float* ab = attn + b * GAT_K * GAT_K;
  const float* xb = x + b * GAT_F * GAT_K;

  v8f c = {};
  for (int j0 = 0; j0 < GAT_K; j0 += 4) {
    v2f av, bv;
    av.x = ab[(i0 + l) * GAT_K + j0 + 2 * hf + 0];
    av.y = ab[(i0 + l) * GAT_K + j0 + 2 * hf + 1];
    bv.x = xb[(f0 + l) * GAT_K + j0 + 2 * hf + 0];
    bv.y = xb[(f0 + l) * GAT_K + j0 + 2 * hf + 1];
    c = __builtin_amdgcn_wmma_f32_16x16x4_f32(false, av, false, bv, (short)0,
                                              c, false, false);
  }
#pragma unroll
  for (int r = 0; r < 8; ++r) {
    const int row = i0 + r + 8 * hf;
    const int col = f0 + l;
    const float z = c[r];
    out[(b * GAT_K + row) * GAT_F + col] = 1.0f / (1.0f + __expf(-z));
  }
}

// ---------------------------------------------------------------------------
extern "C" void kernel_launch(void* const* d_in, const int* in_sizes, int n_in,
                              void* d_out, int out_size, void* d_ws, size_t ws_size,
                              hipStream_t stream) {
  (void)in_sizes; (void)n_in; (void)out_size; (void)ws_size;
  const float* x    = (const float*)d_in[0];   // (16,128,128)
  const float* w    = (const float*)d_in[1];   // (256,256)
  const float* lb   = (const float*)d_in[2];   // (256)
  const float* av   = (const float*)d_in[3];   // (256,1)
  const float* bias = (const float*)d_in[4];   // (128,128)
  float* out = (float*)d_out;                  // (16,128,128)

  float* s1   = (float*)d_ws;                          // 2 MB
  float* s2   = s1 + GAT_B * GAT_K * GAT_H;            // 2 MB
  float* attn = s2 + GAT_B * GAT_K * GAT_H;            // 1 MB

  gat_qk_gemm<<<GAT_B * (GAT_K / 16) * (GAT_H / 16), 32, 0, stream>>>(x, w, s1, s2);
  gat_attn  <<<GAT_B * (GAT_K / 8),               256, 0, stream>>>(s1, s2, lb, av, bias, attn);
  gat_out   <<<GAT_B * (GAT_K / 16) * (GAT_F / 16), 32, 0, stream>>>(attn, x, out);
}